// SparseAutoencoder_42949672960454
// MI455X (gfx1250) — compile-verified
//
#include <hip/hip_runtime.h>
#include <hip/hip_bf16.h>

// ---------------------------------------------------------------------------
// Sparse autoencoder forward for MI455X (gfx1250, wave32, WMMA).
// B=4096, D=1024, M=16384, K=32, MULTI_K=128, AUX_K=64, DEAD_THRESH=256
// ---------------------------------------------------------------------------

#define SAE_B 4096
#define SAE_D 1024
#define SAE_M 16384
#define SAE_K 32
#define SAE_MK 128
#define SAE_AK 64
#define SAE_DEAD 256

typedef __attribute__((ext_vector_type(2))) float v2f;
typedef __attribute__((ext_vector_type(8))) float v8f;

// ---------------------------------------------------------------------------
// Kernel 1: xc = x - input_bias  (B x D), float4 vectorized
// ---------------------------------------------------------------------------
__global__ __launch_bounds__(256) void sae_xc_kernel(
    const float* __restrict__ x, const float* __restrict__ ibias,
    float* __restrict__ xc)
{
    const int i = blockIdx.x * 256 + threadIdx.x;      // float4 index
    const float4 xv = ((const float4*)x)[i];
    const float4 bv = ((const float4*)ibias)[i & (SAE_D / 4 - 1)];
    float4 r;
    r.x = xv.x - bv.x; r.y = xv.y - bv.y;
    r.z = xv.z - bv.z; r.w = xv.w - bv.w;
    ((float4*)xc)[i] = r;
}

// ---------------------------------------------------------------------------
// Kernel 2: encoder GEMM via V_WMMA_F32_16X16X4_F32.
//   pre_act[b,m] = sum_d xc[b,d] * W_enc[m,d] + nbias[m]
//   f_full = relu(pre_act)
// Workgroup tile 128(b) x 128(m); 8 waves in a 2x4 grid, each wave 64x32
// (4x2 accumulator tiles).  KC=16 LDS staging, row stride 20 floats
// (16B-aligned fills, conflict-free fragment reads).
// ---------------------------------------------------------------------------
__global__ __launch_bounds__(256) void sae_enc_gemm_kernel(
    const float* __restrict__ xc, const float* __restrict__ W_enc,
    const float* __restrict__ nbias,
    float* __restrict__ pre_act, float* __restrict__ f_full)
{
    constexpr int KC = 16, LSTR = 20;
    __shared__ float As[128 * LSTR];   // [b][k]
    __shared__ float Bs[128 * LSTR];   // [m][k]

    const int tid  = threadIdx.x;
    const int lane = tid & 31;
    const int wv   = tid >> 5;
    const int b0   = blockIdx.y * 128;
    const int m0   = blockIdx.x * 128;
    const int wb   = (wv & 1) * 64;    // wave b-offset inside block
    const int wm   = (wv >> 1) * 32;   // wave m-offset inside block

    const int lrow  = tid >> 1;        // 0..127 : row loaded by this thread
    const int lkoff = (tid & 1) * 8;   // 0 or 8 : k offset (8 floats each)

    v8f acc[4][2];
#pragma unroll
    for (int tb = 0; tb < 4; ++tb)
#pragma unroll
        for (int tm = 0; tm < 2; ++tm)
            acc[tb][tm] = (v8f){0.f, 0.f, 0.f, 0.f, 0.f, 0.f, 0.f, 0.f};

    const int fr = lane & 15;          // fragment row/col
    const int kh = (lane >> 4) * 2;    // lanes 0-15 -> k{0,1}, 16-31 -> k{2,3}

    for (int ks = 0; ks < SAE_D; ks += KC) {
        const float4 a0 = *(const float4*)&xc[(size_t)(b0 + lrow) * SAE_D + ks + lkoff];
        const float4 a1 = *(const float4*)&xc[(size_t)(b0 + lrow) * SAE_D + ks + lkoff + 4];
        const float4 w0 = *(const float4*)&W_enc[(size_t)(m0 + lrow) * SAE_D + ks + lkoff];
        const float4 w1 = *(const float4*)&W_enc[(size_t)(m0 + lrow) * SAE_D + ks + lkoff + 4];
        __syncthreads();                     // previous stage fully consumed
        *(float4*)&As[lrow * LSTR + lkoff]     = a0;
        *(float4*)&As[lrow * LSTR + lkoff + 4] = a1;
        *(float4*)&Bs[lrow * LSTR + lkoff]     = w0;
        *(float4*)&Bs[lrow * LSTR + lkoff + 4] = w1;
        __syncthreads();                     // stage visible

#pragma unroll
        for (int kk = 0; kk < KC; kk += 4) {
            v2f af[4], bf[2];
#pragma unroll
            for (int tb = 0; tb < 4; ++tb)
                af[tb] = *(const v2f*)&As[(wb + tb * 16 + fr) * LSTR + kk + kh];
#pragma unroll
            for (int tm = 0; tm < 2; ++tm)
                bf[tm] = *(const v2f*)&Bs[(wm + tm * 16 + fr) * LSTR + kk + kh];
#pragma unroll
            for (int tb = 0; tb < 4; ++tb)
#pragma unroll
                for (int tm = 0; tm < 2; ++tm)
                    acc[tb][tm] = __builtin_amdgcn_wmma_f32_16x16x4_f32(
                        false, af[tb], false, bf[tm],
                        (short)0, acc[tb][tm], false, false);
        }
    }

    // Epilogue: C layout — VGPR r, lanes0-15: row r, lanes16-31: row 8+r.
    const int fn  = lane & 15;
    const int f8  = (lane >> 4) * 8;
#pragma unroll
    for (int tm = 0; tm < 2; ++tm) {
        const int m  = m0 + wm + tm * 16 + fn;
        const float nb = nbias[m];
#pragma unroll
        for (int tb = 0; tb < 4; ++tb) {
#pragma unroll
            for (int r = 0; r < 8; ++r) {
                const int b = b0 + wb + tb * 16 + f8 + r;
                const float val = acc[tb][tm][r] + nb;
                pre_act[(size_t)b * SAE_M + m] = val;
                f_full[(size_t)b * SAE_M + m]  = fmaxf(val, 0.0f);
            }
        }
    }
}

// ---------------------------------------------------------------------------
// Kernel 3: new_steps = steps + 1
// ---------------------------------------------------------------------------
__global__ __launch_bounds__(256) void sae_steps_kernel(
    const int* __restrict__ steps, int* __restrict__ new_steps)
{
    const int i = blockIdx.x * 256 + threadIdx.x;
    new_steps[i] = steps[i] + 1;
}

// ---------------------------------------------------------------------------
// Top-k per row (block of 256 threads per row).
// Radix-histogram threshold (11-bit ordered-float bins) + candidate
// compaction + bitonic sort on (key<<32 | ~index) -> descending value,
// ascending index tie-break, matching jax.lax.top_k.
// MODE 0: top-128 of f_full -> topk outputs, activations scatter, multik ws,
//         zero new_steps at fired indices.
// MODE 1: top-64 of pre_act * (new_steps > DEAD_THRESH) -> aux outputs.
// ---------------------------------------------------------------------------
__device__ __forceinline__ unsigned sae_fkey(float f) {
    unsigned u = __float_as_uint(f);
    return (u & 0x80000000u) ? ~u : (u | 0x80000000u);
}
__device__ __forceinline__ float sae_kval(unsigned k) {
    unsigned u = (k & 0x80000000u) ? (k & 0x7FFFFFFFu) : ~k;
    return __uint_as_float(u);
}

template <int MODE, int NSEL>
__global__ __launch_bounds__(256) void sae_topk_kernel(
    const float* __restrict__ src,       // B x M
    int* __restrict__ new_steps,         // read (MODE 1) / scatter-zero (MODE 0)
    float* __restrict__ topk_val, int* __restrict__ topk_idx,
    float* __restrict__ act,
    float* __restrict__ mk_val, int* __restrict__ mk_idx,
    float* __restrict__ aux_val, int* __restrict__ aux_idx)
{
    constexpr int CAP = 1024;
    __shared__ unsigned hist[2048];
    __shared__ unsigned chunk[256];
    __shared__ unsigned long long cand[CAP];
    __shared__ unsigned s_cnt, s_T;

    const int row = blockIdx.x;
    const int tid = threadIdx.x;
    const float* rp = src + (size_t)row * SAE_M;

    for (int i = tid; i < 2048; i += 256) hist[i] = 0u;
    if (tid == 0) s_cnt = 0u;
    __syncthreads();

    // Pass 1: histogram of ordered keys (top 11 bits).
    for (int i = tid; i < SAE_M; i += 256) {
        float v = rp[i];
        if (MODE == 1) v = (new_steps[i] > SAE_DEAD) ? v : 0.0f;
        atomicAdd(&hist[sae_fkey(v) >> 21], 1u);
    }
    __syncthreads();

    unsigned cs = 0;
#pragma unroll
    for (int j = 0; j < 8; ++j) cs += hist[tid * 8 + j];
    chunk[tid] = cs;
    __syncthreads();

    if (tid == 0) {
        unsigned run = 0; int T = 0;
        for (int c = 255; c >= 0; --c) {
            if (run + chunk[c] >= (unsigned)NSEL) {
                for (int b = c * 8 + 7; b >= c * 8; --b) {
                    run += hist[b];
                    if (run >= (unsigned)NSEL) { T = b; break; }
                }
                break;
            }
            run += chunk[c];
        }
        s_T = (unsigned)T;
    }
    __syncthreads();
    const unsigned T = s_T;

    // Pass 2: compact candidates (all elements in bins >= T).
    for (int i = tid; i < SAE_M; i += 256) {
        float v = rp[i];
        if (MODE == 1) v = (new_steps[i] > SAE_DEAD) ? v : 0.0f;
        const unsigned k = sae_fkey(v);
        if ((k >> 21) >= T) {
            const unsigned p = atomicAdd(&s_cnt, 1u);
            if (p < CAP)
                cand[p] = ((unsigned long long)k << 32) | (unsigned)(~(unsigned)i);
        }
    }
    __syncthreads();
    unsigned cnt = s_cnt; if (cnt > CAP) cnt = CAP;
    for (int i = tid; i < CAP; i += 256)
        if (i >= (int)cnt) cand[i] = 0ull;
    __syncthreads();

    // Bitonic sort, descending.
    for (int k2 = 2; k2 <= CAP; k2 <<= 1) {
        for (int j = k2 >> 1; j > 0; j >>= 1) {
            for (int i = tid; i < CAP; i += 256) {
                const int ix = i ^ j;
                if (ix > i) {
                    const unsigned long long a = cand[i], b = cand[ix];
                    const bool asc = (i & k2) != 0;
                    if (asc ? (a > b) : (a < b)) { cand[i] = b; cand[ix] = a; }
                }
            }
            __syncthreads();
        }
    }

    if (MODE == 0) {
        // zero entire activations row, then scatter top-32
        for (int i = tid; i < SAE_M; i += 256)
            act[(size_t)row * SAE_M + i] = 0.0f;
        __syncthreads();
        if (tid < SAE_MK) {
            const unsigned long long p = cand[tid];
            const unsigned k = (unsigned)(p >> 32);
            const int idx = (int)(~(unsigned)p);
            const float v = sae_kval(k);
            mk_val[row * SAE_MK + tid] = v;
            mk_idx[row * SAE_MK + tid] = idx;
            if (tid < SAE_K) {
                topk_val[row * SAE_K + tid] = v;
                topk_idx[row * SAE_K + tid] = idx;
                act[(size_t)row * SAE_M + idx] = v;
                new_steps[idx] = 0;   // benign cross-block race: all write 0
            }
        }
    } else {
        if (tid < SAE_AK) {
            const unsigned long long p = cand[tid];
            const unsigned k = (unsigned)(p >> 32);
            const int idx = (int)(~(unsigned)p);
            const float v = sae_kval(k);
            aux_val[row * SAE_AK + tid] = fmaxf(v, 0.0f);
            aux_idx[row * SAE_AK + tid] = idx;
        }
    }
}

// ---------------------------------------------------------------------------
// Kernel 5: sparse decode.  One block per row; thread t owns d = t + 256*q.
//   recon[b,d]    = ibias[d] + sum_{j<32}  v_j * W_dec[d, idx_j]
//   mk_recon[b,d] = ibias[d] + sum_{j<128} v_j * W_dec[d, idx_j]
// W_dec column gathers are stride-M but W_dec (64 MB) is L2-resident (192 MB).
// ---------------------------------------------------------------------------
__global__ __launch_bounds__(256) void sae_decode_kernel(
    const float* __restrict__ mk_val, const int* __restrict__ mk_idx,
    const float* __restrict__ W_dec, const float* __restrict__ ibias,
    float* __restrict__ recon, float* __restrict__ mk_recon)
{
    __shared__ float sv[SAE_MK];
    __shared__ int   si[SAE_MK];
    const int row = blockIdx.x, tid = threadIdx.x;
    if (tid < SAE_MK) {
        sv[tid] = mk_val[row * SAE_MK + tid];
        si[tid] = mk_idx[row * SAE_MK + tid];
    }
    __syncthreads();

    float a_top[4], a_mk[4];
#pragma unroll
    for (int q = 0; q < 4; ++q) {
        const float b = ibias[tid + 256 * q];
        a_top[q] = b; a_mk[q] = b;
    }
    for (int j = 0; j < SAE_K; ++j) {
        const float v = sv[j]; const int m = si[j];
#pragma unroll
        for (int q = 0; q < 4; ++q) {
            const float w = W_dec[(size_t)(tid + 256 * q) * SAE_M + m];
            a_top[q] += v * w;
            a_mk[q]  += v * w;
        }
    }
    for (int j = SAE_K; j < SAE_MK; ++j) {
        const float v = sv[j]; const int m = si[j];
#pragma unroll
        for (int q = 0; q < 4; ++q)
            a_mk[q] += v * W_dec[(size_t)(tid + 256 * q) * SAE_M + m];
    }
#pragma unroll
    for (int q = 0; q < 4; ++q) {
        recon[(size_t)row * SAE_D + tid + 256 * q]    = a_top[q];
        mk_recon[(size_t)row * SAE_D + tid + 256 * q] = a_mk[q];
    }
}

// ---------------------------------------------------------------------------
// Launcher
// ---------------------------------------------------------------------------
extern "C" void kernel_launch(void* const* d_in, const int* in_sizes, int n_in,
                              void* d_out, int out_size, void* d_ws, size_t ws_size,
                              hipStream_t stream)
{
    (void)in_sizes; (void)n_in; (void)out_size; (void)ws_size;

    const float* x     = (const float*)d_in[0];
    const float* W_enc = (const float*)d_in[1];
    const float* W_dec = (const float*)d_in[2];
    const float* ibias = (const float*)d_in[3];
    const float* nbias = (const float*)d_in[4];
    const int*   steps = (const int*)d_in[5];

    float* out = (float*)d_out;
    // output layout (flat, in reference return order; all 4-byte elements)
    float* o_recon = out;                                   // B*D
    float* o_act   = out + (size_t)4194304;                 // B*M
    int*   o_tidx  = (int*)(out + (size_t)71303168);        // B*K (int32)
    float* o_tval  = out + (size_t)71434240;                // B*K
    float* o_mkrec = out + (size_t)71565312;                // B*D
    int*   o_aidx  = (int*)(out + (size_t)75759616);        // B*AUX_K (int32)
    float* o_aval  = out + (size_t)76021760;                // B*AUX_K
    float* o_ff    = out + (size_t)76283904;                // B*M
    int*   o_steps = (int*)(out + (size_t)143392768);       // M (int32)

    // workspace: pre_act (B*M f32) | xc (B*D f32) | mk_val (B*128) | mk_idx
    float* ws_pre = (float*)d_ws;
    float* ws_xc  = ws_pre + (size_t)SAE_B * SAE_M;
    float* ws_mkv = ws_xc + (size_t)SAE_B * SAE_D;
    int*   ws_mki = (int*)(ws_mkv + (size_t)SAE_B * SAE_MK);

    // 1) xc = x - input_bias
    sae_xc_kernel<<<(SAE_B * SAE_D) / (256 * 4), 256, 0, stream>>>(x, ibias, ws_xc);

    // 2) encoder GEMM (WMMA f32): pre_act -> ws, f_full -> out
    sae_enc_gemm_kernel<<<dim3(SAE_M / 128, SAE_B / 128), 256, 0, stream>>>(
        ws_xc, W_enc, nbias, ws_pre, o_ff);

    // 3) new_steps = steps + 1
    sae_steps_kernel<<<SAE_M / 256, 256, 0, stream>>>(steps, o_steps);

    // 4) top-128 per row: topk outputs, activations, multik pairs, step reset
    sae_topk_kernel<0, SAE_MK><<<SAE_B, 256, 0, stream>>>(
        o_ff, o_steps, o_tval, o_tidx, o_act, ws_mkv, ws_mki, o_aval, o_aidx);

    // 5) sparse decode: reconstruction + multik_reconstruction
    sae_decode_kernel<<<SAE_B, 256, 0, stream>>>(
        ws_mkv, ws_mki, W_dec, ibias, o_recon, o_mkrec);

    // 6) aux top-64 over dead-masked pre-activations
    sae_topk_kernel<1, SAE_AK><<<SAE_B, 256, 0, stream>>>(
        ws_pre, o_steps, o_tval, o_tidx, o_act, ws_mkv, ws_mki, o_aval, o_aidx);
}